// RCK_model_24129126269163
// MI455X (gfx1250) — compile-verified
//
#include <hip/hip_runtime.h>

// ---------------------------------------------------------------------------
// Block-tridiagonal solve (T=524288, M=4) via Block Cyclic Reduction.
// Forward reduction levels use V_WMMA_F32_16X16X4_F32: each wave batches 4
// independent 4x4 matmuls into one 16x16x4 WMMA (diagonal blocks of D).
// ---------------------------------------------------------------------------

typedef float v2f __attribute__((ext_vector_type(2)));
typedef float v8f __attribute__((ext_vector_type(8)));

static constexpr int TT  = 524288;   // horizon
static constexpr int MN1 = TT / 2;   // compacted ws rows after level 0
#define WV 8                         // waves per 256-thread block

__device__ __forceinline__ int imin(int a, int b) { return a < b ? a : b; }
__device__ __forceinline__ int imax(int a, int b) { return a > b ? a : b; }

__device__ __forceinline__ void lds_fence() {
    __asm__ volatile("s_wait_dscnt 0x0" ::: "memory");
}

// 4x4 inverse via adjugate (row-major). Safe for diagonally dominant blocks.
__device__ __forceinline__ void inv4(const float* m, float* o) {
    float s0 = m[0]*m[5]  - m[4]*m[1];
    float s1 = m[0]*m[6]  - m[4]*m[2];
    float s2 = m[0]*m[7]  - m[4]*m[3];
    float s3 = m[1]*m[6]  - m[5]*m[2];
    float s4 = m[1]*m[7]  - m[5]*m[3];
    float s5 = m[2]*m[7]  - m[6]*m[3];
    float c5 = m[10]*m[15] - m[14]*m[11];
    float c4 = m[9]*m[15]  - m[13]*m[11];
    float c3 = m[9]*m[14]  - m[13]*m[10];
    float c2 = m[8]*m[15]  - m[12]*m[11];
    float c1 = m[8]*m[14]  - m[12]*m[10];
    float c0 = m[8]*m[13]  - m[12]*m[9];
    float det = s0*c5 - s1*c4 + s2*c3 + s3*c2 - s4*c1 + s5*c0;
    float id = 1.0f / det;
    o[0]  = ( m[5]*c5 - m[6]*c4 + m[7]*c3) * id;
    o[1]  = (-m[1]*c5 + m[2]*c4 - m[3]*c3) * id;
    o[2]  = ( m[13]*s5 - m[14]*s4 + m[15]*s3) * id;
    o[3]  = (-m[9]*s5 + m[10]*s4 - m[11]*s3) * id;
    o[4]  = (-m[4]*c5 + m[6]*c2 - m[7]*c1) * id;
    o[5]  = ( m[0]*c5 - m[2]*c2 + m[3]*c1) * id;
    o[6]  = (-m[12]*s5 + m[14]*s2 - m[15]*s1) * id;
    o[7]  = ( m[8]*s5 - m[10]*s2 + m[11]*s1) * id;
    o[8]  = ( m[4]*c4 - m[5]*c2 + m[7]*c0) * id;
    o[9]  = (-m[0]*c4 + m[1]*c2 - m[3]*c0) * id;
    o[10] = ( m[12]*s4 - m[13]*s2 + m[15]*s0) * id;
    o[11] = (-m[8]*s4 + m[9]*s2 - m[11]*s0) * id;
    o[12] = (-m[4]*c3 + m[5]*c1 - m[6]*c0) * id;
    o[13] = ( m[0]*c3 - m[1]*c1 + m[2]*c0) * id;
    o[14] = (-m[12]*s3 + m[13]*s1 - m[14]*s0) * id;
    o[15] = ( m[8]*s3 - m[9]*s1 + m[10]*s0) * id;
}

// D = A(16x4) x B(4x16) + C : 4 independent 4x4 matmuls in the diag blocks.
__device__ __forceinline__ v8f wmma4(v2f a, v2f b, v8f c) {
    return __builtin_amdgcn_wmma_f32_16x16x4_f32(false, a, false, b,
                                                 (short)0, c, false, false);
}

// --- WMMA operand builders. LDS slots hold 4 row-major 4x4 matrices (64 f32).
// A-operand (16x4 f32): lanes 0-15 M=lane K={0,1}; lanes 16-31 M=lane-16 K={2,3}.
__device__ __forceinline__ v2f ld_aop_lds(const float* S, int lane) {
    int m = lane & 15, r = m >> 2, mm = m & 3, kb = (lane >> 4) * 2;
    const float* p = S + r * 16 + mm * 4 + kb;
    v2f v; v.x = p[0]; v.y = p[1]; return v;
}
__device__ __forceinline__ v2f ld_aop_g(const float* G, int base, int stride,
                                        int maxRow, int lane) {
    int m = lane & 15, r = m >> 2, mm = m & 3, kb = (lane >> 4) * 2;
    int g = imin(imax(base + stride * r, 0), maxRow);
    const float* p = G + (size_t)g * 16 + mm * 4 + kb;
    v2f v; v.x = p[0]; v.y = p[1]; return v;
}
// B-operand (4x16 f32): VGPR v, lane L -> row = v + 2*(L>>4), col = L&15.
__device__ __forceinline__ v2f ld_bop_lds(const float* S, int lane) {
    int col = lane & 15, c = col >> 2, nn = col & 3, k0 = (lane >> 4) * 2;
    v2f v;
    v.x = S[c * 16 + k0 * 4 + nn];
    v.y = S[c * 16 + (k0 + 1) * 4 + nn];
    return v;
}
__device__ __forceinline__ v2f ld_bop_g(const float* G, int base, int stride,
                                        int maxRow, int cZeroRow, int lane) {
    int col = lane & 15, c = col >> 2, nn = col & 3, k0 = (lane >> 4) * 2;
    int g = imin(imax(base + stride * c, 0), maxRow);
    float x0 = G[(size_t)g * 16 + k0 * 4 + nn];
    float x1 = G[(size_t)g * 16 + (k0 + 1) * 4 + nn];
    if (g == cZeroRow) { x0 = 0.f; x1 = 0.f; }
    v2f v; v.x = x0; v.y = x1; return v;
}

// Each lane of a D accumulator owns (statically) either rows c[0..3] or
// c[4..7] of its diagonal block -- select branchlessly.
struct DiagSel { bool ok; bool hi; int r; int nn; };
__device__ __forceinline__ DiagSel diag_sel(int lane) {
    int col = lane & 15;
    DiagSel s;
    s.r  = col >> 2;
    s.nn = col & 3;
    int vbase = 4 * s.r - (lane >> 4) * 8;   // 0 or 4 when this lane holds data
    s.ok = (vbase == 0) | (vbase == 4);
    s.hi = (vbase == 4);
    return s;
}
__device__ __forceinline__ float dsel(v8f c, bool hi, int i) {
    // i is a compile-time constant after unrolling
    return hi ? c[4 + i] : c[i];
}
// Scatter the 4 diagonal 4x4 blocks of a D accumulator into an LDS slot.
__device__ __forceinline__ void st_diag(float* S, v8f c, int lane) {
    DiagSel s = diag_sel(lane);
    float e0 = dsel(c, s.hi, 0), e1 = dsel(c, s.hi, 1);
    float e2 = dsel(c, s.hi, 2), e3 = dsel(c, s.hi, 3);
    if (s.ok) {
        float* p = S + s.r * 16 + s.nn;   // rows mm=0..3 at stride 4 floats
        p[0] = e0; p[4] = e1; p[8] = e2; p[12] = e3;
    }
}

// ---------------------------------------------------------------------------
// One BCR reduction level. kept row k: self = selfMul*k, neighbors self +/- nbD,
// output row = outMul*k. Level 0: d_in -> ws (compact). ws levels: in-place.
// No early exit: EXEC stays all-ones at every WMMA; all stores are guarded.
// ---------------------------------------------------------------------------
__global__ __launch_bounds__(256) void bcr_reduce(
    const float* Ain, const float* Bin, const float* Cin, const float* din,
    float* Aout, float* Bout, float* Cout, float* dout,
    int Nk, int Nin, int selfMul, int nbD, int outMul, int cZeroRow)
{
    __shared__ float lds[WV * 256];
    const int lane = threadIdx.x & 31;
    const int wave = threadIdx.x >> 5;
    float* L = lds + wave * 256;            // [INVBM 64][INVBP 64][ALPHA 64][GAMMA 64]
    const int k0 = (blockIdx.x * WV + wave) * 4;

    const int maxR = Nin - 1;
    const int selfBase = selfMul * k0;

    // --- stage inverses of neighbor B blocks (8 lanes, one 4x4 each) ---
    if (lane < 8) {
        int r = lane & 3, side = lane >> 2;  // side 0: left, 1: right
        int kb = k0 + r;
        int g = selfBase + selfMul * r + (side ? nbD : -nbD);
        bool valid = (kb < Nk) && (side == 1 || kb > 0);
        g = imin(imax(g, 0), maxR);
        float Bm[16], IB[16];
        const float4* p = (const float4*)(Bin + (size_t)g * 16);
#pragma unroll
        for (int i = 0; i < 4; ++i) {
            float4 q = p[i];
            Bm[i*4+0]=q.x; Bm[i*4+1]=q.y; Bm[i*4+2]=q.z; Bm[i*4+3]=q.w;
        }
        inv4(Bm, IB);
        float* slot = L + side * 64 + r * 16;
#pragma unroll
        for (int i = 0; i < 16; ++i) slot[i] = valid ? IB[i] : 0.0f;
    }
    lds_fence();

    // --- alpha = A_i * invB_minus ; gamma = C_i * invB_plus  (2 WMMAs) ---
    v8f zacc = {};
    v8f alpha = wmma4(ld_aop_g(Ain, selfBase, selfMul, maxR, lane),
                      ld_bop_lds(L + 0, lane), zacc);
    v8f gamma = wmma4(ld_aop_g(Cin, selfBase, selfMul, maxR, lane),
                      ld_bop_lds(L + 64, lane), zacc);
    st_diag(L + 128, alpha, lane);
    st_diag(L + 192, gamma, lane);
    lds_fence();

    // --- neighbor products (4 WMMAs) ---
    v2f aAl = ld_aop_lds(L + 128, lane);
    v2f aGa = ld_aop_lds(L + 192, lane);
    const int mBase = selfBase - nbD;
    const int pBase = selfBase + nbD;
    v8f paa = wmma4(aAl, ld_bop_g(Ain, mBase, selfMul, maxR, -1,      lane), zacc);
    v8f pac = wmma4(aAl, ld_bop_g(Cin, mBase, selfMul, maxR, cZeroRow, lane), zacc);
    v8f pga = wmma4(aGa, ld_bop_g(Ain, pBase, selfMul, maxR, -1,      lane), zacc);
    v8f pgc = wmma4(aGa, ld_bop_g(Cin, pBase, selfMul, maxR, cZeroRow, lane), zacc);

    // --- write A' = -alpha*A-, B' = B - alpha*C- - gamma*A+, C' = -gamma*C+ ---
    {
        DiagSel s = diag_sel(lane);
        int kb = k0 + s.r;
        if (s.ok && kb < Nk) {
            size_t sg = (size_t)(selfBase + selfMul * s.r) * 16 + s.nn;
            size_t og = (size_t)(outMul * kb) * 16 + s.nn;
#pragma unroll
            for (int i = 0; i < 4; ++i) {        // mm == i for the owned half
                size_t e = (size_t)i * 4;
                Aout[og + e] = -dsel(paa, s.hi, i);
                Bout[og + e] = Bin[sg + e] - dsel(pac, s.hi, i)
                                           - dsel(pga, s.hi, i);
                Cout[og + e] = -dsel(pgc, s.hi, i);
            }
        }
    }

    // --- d' = d - alpha*d- - gamma*d+  (16 lanes, one element each) ---
    if (lane < 16) {
        int b = lane >> 2, m = lane & 3;
        int kb = k0 + b;
        if (kb < Nk) {
            int sg = selfBase + selfMul * b;
            int og = outMul * kb;
            int mg = imin(imax(sg - nbD, 0), maxR);
            int pg = imin(sg + nbD, maxR);
            float acc = din[(size_t)sg * 4 + m];
#pragma unroll
            for (int j = 0; j < 4; ++j)
                acc -= L[128 + b * 16 + m * 4 + j] * din[(size_t)mg * 4 + j];
#pragma unroll
            for (int j = 0; j < 4; ++j)
                acc -= L[192 + b * 16 + m * 4 + j] * din[(size_t)pg * 4 + j];
            dout[(size_t)og * 4 + m] = acc;
        }
    }
}

// 1x1 block system at the top of the reduction tree: x0 = B^-1 d.
__global__ void bcr_top_solve(const float* Bw, const float* dw, float* x) {
    if (threadIdx.x != 0 || blockIdx.x != 0) return;
    float B[16], IB[16];
#pragma unroll
    for (int i = 0; i < 16; ++i) B[i] = Bw[i];
    inv4(B, IB);
#pragma unroll
    for (int m = 0; m < 4; ++m) {
        float a = 0.f;
#pragma unroll
        for (int j = 0; j < 4; ++j) a += IB[m * 4 + j] * dw[j];
        x[m] = a;
    }
}

// Back-substitution level: solve rows self=(2k+1)*s using known x at self +/- s.
// x position of row j is j*xMul (xMul=2 for compacted ws rows, 1 for level 0).
__global__ __launch_bounds__(256) void bcr_backsub(
    const float* Ain, const float* Bin, const float* Cin, const float* din,
    float* x, int Nk, int Nin, int s, int xMul)
{
    int k = blockIdx.x * blockDim.x + threadIdx.x;
    if (k >= Nk) return;
    int self = (2 * k + 1) * s;
    float B[16], IB[16];
    const float4* bp = (const float4*)(Bin + (size_t)self * 16);
#pragma unroll
    for (int i = 0; i < 4; ++i) {
        float4 q = bp[i];
        B[i*4+0]=q.x; B[i*4+1]=q.y; B[i*4+2]=q.z; B[i*4+3]=q.w;
    }
    inv4(B, IB);
    float r[4];
#pragma unroll
    for (int m = 0; m < 4; ++m) r[m] = din[(size_t)self * 4 + m];
    const float* xm = x + (size_t)(self - s) * xMul * 4;
#pragma unroll
    for (int m = 0; m < 4; ++m)
#pragma unroll
        for (int j = 0; j < 4; ++j)
            r[m] -= Ain[(size_t)self * 16 + m * 4 + j] * xm[j];
    if (self + s < Nin) {
        const float* xp = x + (size_t)(self + s) * xMul * 4;
#pragma unroll
        for (int m = 0; m < 4; ++m)
#pragma unroll
            for (int j = 0; j < 4; ++j)
                r[m] -= Cin[(size_t)self * 16 + m * 4 + j] * xp[j];
    }
    float* xo = x + (size_t)self * xMul * 4;
#pragma unroll
    for (int m = 0; m < 4; ++m) {
        float a = 0.f;
#pragma unroll
        for (int j = 0; j < 4; ++j) a += IB[m * 4 + j] * r[j];
        xo[m] = a;
    }
}

extern "C" void kernel_launch(void* const* d_in, const int* in_sizes, int n_in,
                              void* d_out, int out_size, void* d_ws, size_t ws_size,
                              hipStream_t stream) {
    const float* A = (const float*)d_in[0];
    const float* B = (const float*)d_in[1];
    const float* C = (const float*)d_in[2];
    const float* d = (const float*)d_in[3];
    float* x  = (float*)d_out;
    float* ws = (float*)d_ws;

    float* Aw = ws;
    float* Bw = ws + (size_t)MN1 * 16;
    float* Cw = ws + (size_t)MN1 * 32;
    float* dw = ws + (size_t)MN1 * 48;   // total 52*MN1 floats (~54.5 MB)

    const int ROWS_PER_BLOCK = WV * 4;   // 32 kept rows per 256-thread block

    // Forward: level 0 (reads immutable inputs, writes compacted ws; C[T-1]:=0)
    {
        int Nk = MN1;
        dim3 g((Nk + ROWS_PER_BLOCK - 1) / ROWS_PER_BLOCK), b(256);
        bcr_reduce<<<g, b, 0, stream>>>(A, B, C, d, Aw, Bw, Cw, dw,
                                        Nk, TT, 2, 1, 1, TT - 1);
    }
    // Forward: ws levels, in place (eliminated rows keep their level values)
    for (int s = 1; s < MN1; s <<= 1) {
        int Nk = MN1 / (2 * s);
        dim3 g((Nk + ROWS_PER_BLOCK - 1) / ROWS_PER_BLOCK), b(256);
        bcr_reduce<<<g, b, 0, stream>>>(Aw, Bw, Cw, dw, Aw, Bw, Cw, dw,
                                        Nk, MN1, 2 * s, s, 2 * s, -1);
    }
    // Top of the tree: x(t=0)
    bcr_top_solve<<<dim3(1), dim3(32), 0, stream>>>(Bw, dw, x);
    // Backward: ws levels (even t), then level 0 (odd t, reads inputs)
    for (int s = MN1 / 2; s >= 1; s >>= 1) {
        int Nk = MN1 / (2 * s);
        dim3 g((Nk + 255) / 256), b(256);
        bcr_backsub<<<g, b, 0, stream>>>(Aw, Bw, Cw, dw, x, Nk, MN1, s, 2);
    }
    {
        int Nk = TT / 2;
        dim3 g((Nk + 255) / 256), b(256);
        bcr_backsub<<<g, b, 0, stream>>>(A, B, C, d, x, Nk, TT, 1, 1);
    }
}